// BilinearFusion_86681029967956
// MI455X (gfx1250) — compile-verified
//
#include <hip/hip_runtime.h>
#include <stdint.h>

#define BATCH    4096
#define IMG_D    2048
#define TXT_D    768
#define OUT_D    8192
#define OUT_MASK 8191
#define LN_EPS   1e-5f
#define BLOCK    256
#define KPT      (OUT_D / BLOCK)   /* 32 outputs per thread */

typedef unsigned int u32;
typedef __attribute__((ext_vector_type(4))) unsigned int v4u32;
typedef __attribute__((ext_vector_type(4))) int          v4i32;
typedef __attribute__((ext_vector_type(8))) int          v8i32;

// LDS layout (in floats): sk1 | xrow | yrow | interleaved (c,h2) pairs | reduction
#define SK1_OFF   0
#define XROW_OFF  (OUT_D)                        /* 8192  -> byte 32768 */
#define YROW_OFF  (OUT_D + IMG_D)                /* 10240 -> byte 40960 */
#define CS_OFF    (OUT_D + IMG_D + TXT_D)        /* 11008 -> byte 44032 (8B aligned) */
#define RED_OFF   (OUT_D + IMG_D + 3 * TXT_D)    /* 12544 */
#define SMEM_FLOATS (RED_OFF + 16)               /* 12560 floats = 50240 bytes */

#if defined(__HIP_DEVICE_COMPILE__) && __has_builtin(__builtin_amdgcn_tensor_load_to_lds)
#define HAVE_TDM 1
// Issue one TDM descriptor copying a contiguous row of `nelem` f32 from global
// memory into LDS at byte offset `lds_byte_off`.  D# layout per CDNA5 ISA §8.3/8.4:
//   group0: count=1 | lds_addr | global_addr[56:0] | type=2
//   group1: data_size=4B, tensor_dim0 = tile_dim0 = nelem, tensor_dim1 = tile_dim1 = 1,
//           tensor_dim0_stride = nelem; groups 2/3 NULL (<=2D tensor)
__device__ __forceinline__ void tdm_load_row(const float* src, u32 lds_byte_off, u32 nelem)
{
    const uint64_t ga = (uint64_t)(const void*)src;
    v4u32 g0;
    g0.x = 1u;                                              // count=1 (valid user D#)
    g0.y = lds_byte_off;                                    // lds_addr (bytes)
    g0.z = (u32)(ga & 0xffffffffu);                         // global_addr[31:0]
    g0.w = (u32)((ga >> 32) & 0x01ffffffu) | 0x80000000u;   // global_addr[56:32] | type=2
    v8i32 g1;
    g1[0] = 0x00020000;                                     // workgroup_mask=0, data_size=2 (4B)
    g1[1] = (int)((nelem & 0xffffu) << 16);                 // tensor_dim0[15:0] in bits[63:48]
    g1[2] = (int)((nelem >> 16) | (1u << 16));              // tensor_dim0[31:16] | tensor_dim1=1
    g1[3] = (int)((nelem & 0xffffu) << 16);                 // tensor_dim1 hi=0 | tile_dim0=nelem
    g1[4] = 1;                                              // tile_dim1=1, tile_dim2=0
    g1[5] = (int)nelem;                                     // tensor_dim0_stride[31:0]
    g1[6] = (int)((nelem & 0xffffu) << 16);                 // stride0 hi=0 | dim1_stride lo
    g1[7] = 0;
    v4i32 gz = {0, 0, 0, 0};
#if __clang_major__ >= 23
    v8i32 gz8 = {0, 0, 0, 0, 0, 0, 0, 0};
    __builtin_amdgcn_tensor_load_to_lds(g0, g1, gz, gz, gz8, 0);
#else
    __builtin_amdgcn_tensor_load_to_lds(g0, g1, gz, gz, 0);
#endif
}
#else
#define HAVE_TDM 0
#endif

__global__ __launch_bounds__(BLOCK)
void BilinearFusion_mcb_conv_ln(const float* __restrict__ img,
                                const float* __restrict__ txt,
                                const int*   __restrict__ h1,
                                const int*   __restrict__ h2,
                                const float* __restrict__ s1,
                                const float* __restrict__ s2,
                                const float* __restrict__ gamma,
                                const float* __restrict__ beta,
                                float* __restrict__ out)
{
    __shared__ __align__(16) float smem[SMEM_FLOATS];
    float* sk1  = smem + SK1_OFF;
    float* xrow = smem + XROW_OFF;
    float* yrow = smem + YROW_OFF;
    float* cs   = smem + CS_OFF;
    float* red  = smem + RED_OFF;

    const int tid = threadIdx.x;
    const int b   = blockIdx.x;

    // ---- Phase A: wave 0 DMAs this row's image+text embeddings into LDS via the
    // Tensor Data Mover and drains its own TENSORcnt, while waves 1..7 zero the
    // 32KB sketch accumulator in parallel.  Single guarded region (issue+wait
    // together) keeps the structurizer from exec-guarding the zero-fill loop.
#if HAVE_TDM
    if (tid < 32) {
        tdm_load_row(img + (size_t)b * IMG_D, (u32)(XROW_OFF * 4), IMG_D);
        tdm_load_row(txt + (size_t)b * TXT_D, (u32)(YROW_OFF * 4), TXT_D);
        __builtin_amdgcn_s_wait_tensorcnt(0);   // drains both tensor DMAs (in-order)
    }
#else
    for (int i = tid; i < IMG_D; i += BLOCK) xrow[i] = img[(size_t)b * IMG_D + i];
    for (int j = tid; j < TXT_D; j += BLOCK) yrow[j] = txt[(size_t)b * TXT_D + j];
#endif
    for (int i = tid; i < OUT_D; i += BLOCK) sk1[i] = 0.0f;
    __syncthreads();

    // ---- Phase B: build image count-sketch + cache signed text features ----
    for (int i = tid; i < IMG_D; i += BLOCK) {
        const float v = xrow[i] * s1[i];
        atomicAdd(&sk1[h1[i] & OUT_MASK], v);    // ds_add_f32
    }
    for (int j = tid; j < TXT_D; j += BLOCK) {
        cs[2 * j]             = s2[j] * yrow[j];
        ((int*)cs)[2 * j + 1] = h2[j] & OUT_MASK;
    }
    __syncthreads();

    // ---- Phase C: circular convolution as shifted gather-FMA ----
    // fused[k] = sum_j c[j] * sk1[(k - h2[j]) mod d]; k = tid + 256*m.
    // Consecutive lanes hit consecutive LDS banks -> conflict-free gathers;
    // the (c, shift) pair is one broadcast ds_load_b64 per j.
    float r[KPT];
#pragma unroll
    for (int m = 0; m < KPT; ++m) r[m] = 0.0f;

    for (int j = 0; j < TXT_D; ++j) {
        const float2 p  = *(const float2*)(cs + 2 * j);
        const float  c  = p.x;
        const int    sh = __float_as_int(p.y);
        const int  base = tid - sh;
#pragma unroll
        for (int m = 0; m < KPT; ++m) {
            r[m] = fmaf(c, sk1[(base + (m << 8)) & OUT_MASK], r[m]);
        }
    }

    // ---- Phase D: LayerNorm over d ----
    float sum = 0.0f, ssq = 0.0f;
#pragma unroll
    for (int m = 0; m < KPT; ++m) { sum += r[m]; ssq = fmaf(r[m], r[m], ssq); }
#pragma unroll
    for (int off = 16; off > 0; off >>= 1) {     // wave32 butterfly reduce
        sum += __shfl_xor(sum, off, 32);
        ssq += __shfl_xor(ssq, off, 32);
    }
    if ((tid & 31) == 0) {
        red[2 * (tid >> 5)]     = sum;
        red[2 * (tid >> 5) + 1] = ssq;
    }
    __syncthreads();
    float ts = 0.0f, tq = 0.0f;
#pragma unroll
    for (int w = 0; w < BLOCK / 32; ++w) { ts += red[2 * w]; tq += red[2 * w + 1]; }
    const float mu  = ts * (1.0f / OUT_D);
    const float inv = rsqrtf(fmaf(-mu, mu, tq * (1.0f / OUT_D)) + LN_EPS);

    float* orow = out + (size_t)b * OUT_D;
#pragma unroll
    for (int m = 0; m < KPT; ++m) {
        const int k = tid + (m << 8);
        orow[k] = fmaf((r[m] - mu) * inv, gamma[k], beta[k]);   // coalesced stores
    }
}

extern "C" void kernel_launch(void* const* d_in, const int* in_sizes, int n_in,
                              void* d_out, int out_size, void* d_ws, size_t ws_size,
                              hipStream_t stream) {
    (void)in_sizes; (void)n_in; (void)d_ws; (void)ws_size; (void)out_size;
    const float* img   = (const float*)d_in[0];
    const float* txt   = (const float*)d_in[1];
    const int*   h1    = (const int*)  d_in[2];
    const int*   h2    = (const int*)  d_in[3];
    const float* s1    = (const float*)d_in[4];
    const float* s2    = (const float*)d_in[5];
    const float* gamma = (const float*)d_in[6];
    const float* beta  = (const float*)d_in[7];
    float* out = (float*)d_out;

    BilinearFusion_mcb_conv_ln<<<dim3(BATCH), dim3(BLOCK), 0, stream>>>(
        img, txt, h1, h2, s1, s2, gamma, beta, out);
}